// Convolve_13297218748814
// MI455X (gfx1250) — compile-verified
//
#include <hip/hip_runtime.h>
#include <math.h>

typedef float v2f __attribute__((ext_vector_type(2)));
typedef float v8f __attribute__((ext_vector_type(8)));

#define NN     10000
#define KNBR   32
#define CC     64
#define HH     64
#define ALPHA  0.3f
#define ROWBLKS ((NN + 15) / 16)   // 625

__device__ __forceinline__ float leaky(float v) {
    return v >= 0.0f ? v : ALPHA * v;
}

// ---------------------------------------------------------------------------
// K1: H_all = leaky(embs @ Q_w + Q_b)   (10000x64)x(64x64), fp32 WMMA 16x16x4
// One wave -> 16 rows x 64 cols (4 accumulator tiles).
// Q_w staged in LDS in K-paired layout: sB[(k>>1)*128 + col*2 + (k&1)], so a
// B fragment {B[k][col], B[k+1][col]} is one aligned ds_load_b64.
// ---------------------------------------------------------------------------
__global__ void k_qgemm(const float* __restrict__ embs,
                        const float* __restrict__ Qw,
                        const float* __restrict__ Qb,
                        float* __restrict__ Hall) {
    __shared__ float sB[CC * HH];                 // 16 KB, paired layout
    for (int i = threadIdx.x; i < CC * HH; i += 256) {
        const int k = i >> 6, col = i & 63;
        sB[(k >> 1) * 128 + col * 2 + (k & 1)] = Qw[i];
    }
    __syncthreads();

    const int wave   = threadIdx.x >> 5;
    const int lane   = threadIdx.x & 31;
    const int rowblk = blockIdx.x * 8 + wave;
    if (rowblk >= ROWBLKS) return;

    const int row0 = rowblk * 16;
    const int l15  = lane & 15;
    const int hsel = lane >> 4;                   // 0: lanes 0-15, 1: lanes 16-31
    const int arow = row0 + l15;

    v8f acc[4] = {};
    #pragma unroll
    for (int kk = 0; kk < CC / 4; ++kk) {
        const int kb   = kk * 4 + 2 * hsel;       // A VGPR0 holds K=kb, VGPR1 K=kb+1
        const int pair = 2 * kk + hsel;           // kb >> 1
        v2f a = *(const v2f*)(embs + arow * CC + kb);
        #pragma unroll
        for (int t = 0; t < 4; ++t) {
            v2f b = *(const v2f*)(sB + pair * 128 + (t * 16 + l15) * 2);
            acc[t] = __builtin_amdgcn_wmma_f32_16x16x4_f32(
                false, a, false, b, (short)0, acc[t], false, false);
        }
    }

    #pragma unroll
    for (int t = 0; t < 4; ++t) {
        const int col  = t * 16 + l15;
        const float bv = Qb[col];
        #pragma unroll
        for (int r = 0; r < 8; ++r) {
            float v = leaky(acc[t][r] + bv);
            Hall[(row0 + r + 8 * hsel) * HH + col] = v;
        }
    }
}

// ---------------------------------------------------------------------------
// K2: weighted neighbor sum over H_all (L2-resident 2.56MB table).
// One wave per node; lane k owns neighbor k's index + weight. Fully unrolled
// broadcast loop -> constant-lane readlane broadcasts, no LDS traffic.
// ---------------------------------------------------------------------------
__global__ void k_wsum(const float* __restrict__ Hall,
                       const float* __restrict__ weights,
                       const int* __restrict__ nbr,
                       float* __restrict__ wsh) {
    const int lane = threadIdx.x & 31;
    const int node = (blockIdx.x * blockDim.x + threadIdx.x) >> 5;
    if (node >= NN) return;

    const int   idx = nbr[node * KNBR + lane];
    const float w   = weights[(size_t)node * NN + (size_t)idx];

    float wsum = w;
    #pragma unroll
    for (int off = 16; off >= 1; off >>= 1)
        wsum += __shfl_xor(wsum, off, 32);

    float ax = 0.0f, ay = 0.0f;
    #pragma unroll
    for (int k = 0; k < KNBR; ++k) {
        const float wk = __shfl(w, k, 32);
        const int   nk = __shfl(idx, k, 32);
        const v2f h = *(const v2f*)(Hall + nk * HH + lane * 2);
        ax += wk * h.x;
        ay += wk * h.y;
    }
    const float inv = 1.0f / (wsum + 1e-6f);
    v2f o; o.x = ax * inv; o.y = ay * inv;
    *(v2f*)(wsh + node * HH + lane * 2) = o;
}

// ---------------------------------------------------------------------------
// K3: out = normalize(leaky([embs | wsh] @ W_w + W_b))   K=128, W_w in LDS
// (same K-paired layout). Row L2-norm computed in-register via width-16
// shfl_xor reductions (each C/D row lives in one VGPR across a 16-lane half).
// ---------------------------------------------------------------------------
__global__ void k_wgemm(const float* __restrict__ embs,
                        const float* __restrict__ wsh,
                        const float* __restrict__ Ww,
                        const float* __restrict__ Wb,
                        float* __restrict__ out) {
    __shared__ float sB[(CC + HH) * HH];          // 32 KB, paired layout
    for (int i = threadIdx.x; i < (CC + HH) * HH; i += 256) {
        const int k = i >> 6, col = i & 63;
        sB[(k >> 1) * 128 + col * 2 + (k & 1)] = Ww[i];
    }
    __syncthreads();

    const int wave   = threadIdx.x >> 5;
    const int lane   = threadIdx.x & 31;
    const int rowblk = blockIdx.x * 8 + wave;
    if (rowblk >= ROWBLKS) return;

    const int row0 = rowblk * 16;
    const int l15  = lane & 15;
    const int hsel = lane >> 4;
    const int arow = row0 + l15;

    v8f acc[4] = {};
    #pragma unroll
    for (int kk = 0; kk < (CC + HH) / 4; ++kk) {
        const int c    = kk * 4 + 2 * hsel;       // concat column (uniform side per wave)
        const int pair = 2 * kk + hsel;
        const float* src = (c < CC) ? (embs + arow * CC + c)
                                    : (wsh + arow * HH + (c - CC));
        v2f a = *(const v2f*)src;
        #pragma unroll
        for (int t = 0; t < 4; ++t) {
            v2f b = *(const v2f*)(sB + pair * 128 + (t * 16 + l15) * 2);
            acc[t] = __builtin_amdgcn_wmma_f32_16x16x4_f32(
                false, a, false, b, (short)0, acc[t], false, false);
        }
    }

    // bias + leaky
    #pragma unroll
    for (int t = 0; t < 4; ++t) {
        const float bv = Wb[t * 16 + l15];
        #pragma unroll
        for (int r = 0; r < 8; ++r)
            acc[t][r] = leaky(acc[t][r] + bv);
    }

    // per-row L2 norm + scale + store
    #pragma unroll
    for (int r = 0; r < 8; ++r) {
        float sq = 0.0f;
        #pragma unroll
        for (int t = 0; t < 4; ++t) sq += acc[t][r] * acc[t][r];
        sq += __shfl_xor(sq, 1, 16);
        sq += __shfl_xor(sq, 2, 16);
        sq += __shfl_xor(sq, 4, 16);
        sq += __shfl_xor(sq, 8, 16);
        const float f = 1.0f / (sqrtf(sq) + 1e-6f);
        const int orow = (row0 + r + 8 * hsel) * HH;
        #pragma unroll
        for (int t = 0; t < 4; ++t)
            out[orow + t * 16 + l15] = acc[t][r] * f;
    }
}

extern "C" void kernel_launch(void* const* d_in, const int* in_sizes, int n_in,
                              void* d_out, int out_size, void* d_ws, size_t ws_size,
                              hipStream_t stream) {
    const float* embs    = (const float*)d_in[0];   // (1, N, C)
    const float* weights = (const float*)d_in[1];   // (N, N)
    const int*   nbr     = (const int*)d_in[2];     // (N, K)
    const float* Qw      = (const float*)d_in[3];   // (C, H)
    const float* Qb      = (const float*)d_in[4];   // (H)
    const float* Ww      = (const float*)d_in[5];   // (C+H, H)
    const float* Wb      = (const float*)d_in[6];   // (H)
    float* out = (float*)d_out;                     // (1, N, H)

    float* Hall = (float*)d_ws;                     // N*H floats
    float* wsh  = Hall + (size_t)NN * HH;           // N*H floats

    const int gemmGrid = (ROWBLKS + 7) / 8;         // 8 waves/block, 1 rowblk/wave
    k_qgemm<<<gemmGrid, 256, 0, stream>>>(embs, Qw, Qb, Hall);
    k_wsum <<<(NN * 32 + 255) / 256, 256, 0, stream>>>(Hall, weights, nbr, wsh);
    k_wgemm<<<gemmGrid, 256, 0, stream>>>(embs, wsh, Ww, Wb, out);
}